// MinkUNetDiff_79173427135030
// MI455X (gfx1250) — compile-verified
//
#include <hip/hip_runtime.h>
#include <math.h>

// ---------------------------------------------------------------------------
// MinkUNetDiff forward on gfx1250 (MI455X).
// All conv / deconv GEMMs run on the WMMA pipe via V_WMMA_F32_16X16X4_F32
// (f32-exact). Wave32: one 32(M-voxels) x 16(N-outch) tile per wave
// (2 M-subtiles sharing each B fragment). Weights pre-packed in
// chunk-interleaved wave order so each wave's B-fragment load is one
// coalesced 256B transaction (32 lanes x b64).
// ---------------------------------------------------------------------------

typedef float v2f __attribute__((ext_vector_type(2)));
typedef float v8f __attribute__((ext_vector_type(8)));

__device__ __forceinline__ v8f wmma_f32_k4(v2f a, v2f b, v8f acc) {
#if defined(__HIP_DEVICE_COMPILE__)
#if __has_builtin(__builtin_amdgcn_wmma_f32_16x16x4_f32)
  return __builtin_amdgcn_wmma_f32_16x16x4_f32(false, a, false, b, (short)0,
                                               acc, false, false);
#else
#error "gfx1250 __builtin_amdgcn_wmma_f32_16x16x4_f32 not available"
#endif
#else
  // Host pass placeholder (never executed; __global__ bodies only run on GPU).
  acc[0] += a.x * b.x + a.y * b.y;
  return acc;
#endif
}

// ---------------------------------------------------------------------------
// Weight repack into wave-coalesced chunk order:
//   wp[(((nt*TAPS + tap)*CH + j)*32 + l)*2 + e] =
//        w[(tap*IC + c)*OC + nt*16 + (l&15)],  c = 4j + 2*(l>>4) + e
// (zero-padded for c >= IC).  CH = ICp/4.  One chunk = 64 floats = the exact
// 32-lane B fragment of V_WMMA_F32_16X16X4_F32.
// ---------------------------------------------------------------------------
__global__ void __launch_bounds__(256) pack_w_kernel(
    const float* __restrict__ w, float* __restrict__ wp, int TAPS, int IC,
    int OC, int ICp) {
  const int CH = ICp >> 2;
  const size_t T = (size_t)(OC >> 4) * TAPS * CH * 64;
  for (size_t i = (size_t)blockIdx.x * blockDim.x + threadIdx.x; i < T;
       i += (size_t)gridDim.x * blockDim.x) {
    const int e = (int)(i & 1);
    const int l = (int)((i >> 1) & 31);
    size_t r = i >> 6;
    const int j = (int)(r % CH);
    r /= CH;
    const int tap = (int)(r % TAPS);
    const int nt = (int)(r / TAPS);
    const int c = 4 * j + 2 * (l >> 4) + e;
    const int oc = nt * 16 + (l & 15);
    wp[i] = (c < IC) ? w[((size_t)tap * IC + c) * OC + oc] : 0.f;
  }
}

// ---------------------------------------------------------------------------
// Generic conv3d: input [G_in^3, IC] (NDHWC, B=1), packed weights (see above),
// output [G_out^3, OC].
// Implicit GEMM: M=2x16 voxels, N=16 out-ch, K chunks of 4 per tap.
// ---------------------------------------------------------------------------
__global__ void __launch_bounds__(256) conv3d_wmma_kernel(
    const float* __restrict__ in, const float* __restrict__ wp,
    float* __restrict__ out, int G_in, int G_out, int IC, int OC, int KS,
    int STRIDE, int PAD) {
  const int lane = threadIdx.x & 31;
  const int wid = threadIdx.x >> 5;
  const int Nvox = G_out * G_out * G_out;
  const int MT = (Nvox + 31) >> 5;
  const int NT = OC >> 4;
  const int tile = blockIdx.x * 8 + wid;
  if (tile >= MT * NT) return;  // wave-uniform exit: EXEC stays all-ones
  const int mt = tile % MT;
  const int nt = tile / MT;

  const int mrow = lane & 15;
  const int hi = lane >> 4;  // 0: K+{0,1}, 1: K+{2,3} (A and B layouts)
  const int oc = nt * 16 + (lane & 15);

  const int ICp = (IC + 3) & ~3;
  const int CH = ICp >> 2;
  const int TAPS = KS * KS * KS;
  const bool icAligned = (IC & 3) == 0;

  // Two M rows handled by this lane (A fragments for two 16-voxel subtiles).
  int v0 = mt * 32 + mrow;
  int v1 = v0 + 16;
  const float mm0 = (v0 < Nvox) ? 1.f : 0.f;
  const float mm1 = (v1 < Nvox) ? 1.f : 0.f;
  if (v0 >= Nvox) v0 = Nvox - 1;
  if (v1 >= Nvox) v1 = Nvox - 1;
  const int GG = G_out * G_out;
  const int z0 = v0 / GG, q0 = v0 - z0 * GG, y0 = q0 / G_out,
            x0 = q0 - (q0 / G_out) * G_out;
  const int z1 = v1 / GG, q1 = v1 - z1 * GG, y1 = q1 / G_out,
            x1 = q1 - (q1 / G_out) * G_out;

  v8f acc0 = {0.f, 0.f, 0.f, 0.f, 0.f, 0.f, 0.f, 0.f};
  v8f acc1 = {0.f, 0.f, 0.f, 0.f, 0.f, 0.f, 0.f, 0.f};

  for (int tap = 0; tap < TAPS; ++tap) {
    const int kd = tap / (KS * KS);
    const int kq = tap - kd * KS * KS;
    const int kh = kq / KS;
    const int kw = kq - kh * KS;

    // Row 0 input coords (clamped address + mask).
    const int iz0 = z0 * STRIDE - PAD + kd;
    const int iy0 = y0 * STRIDE - PAD + kh;
    const int ix0 = x0 * STRIDE - PAD + kw;
    const bool inb0 = ((unsigned)iz0 < (unsigned)G_in) &&
                      ((unsigned)iy0 < (unsigned)G_in) &&
                      ((unsigned)ix0 < (unsigned)G_in);
    const int cz0 = iz0 < 0 ? 0 : (iz0 >= G_in ? G_in - 1 : iz0);
    const int cy0 = iy0 < 0 ? 0 : (iy0 >= G_in ? G_in - 1 : iy0);
    const int cx0 = ix0 < 0 ? 0 : (ix0 >= G_in ? G_in - 1 : ix0);
    const float am0 = inb0 ? mm0 : 0.f;
    const float* ap0 = in + (((size_t)cz0 * G_in + cy0) * G_in + cx0) * IC;

    // Row 1 input coords.
    const int iz1 = z1 * STRIDE - PAD + kd;
    const int iy1 = y1 * STRIDE - PAD + kh;
    const int ix1 = x1 * STRIDE - PAD + kw;
    const bool inb1 = ((unsigned)iz1 < (unsigned)G_in) &&
                      ((unsigned)iy1 < (unsigned)G_in) &&
                      ((unsigned)ix1 < (unsigned)G_in);
    const int cz1 = iz1 < 0 ? 0 : (iz1 >= G_in ? G_in - 1 : iz1);
    const int cy1 = iy1 < 0 ? 0 : (iy1 >= G_in ? G_in - 1 : iy1);
    const int cx1 = ix1 < 0 ? 0 : (ix1 >= G_in ? G_in - 1 : ix1);
    const float am1 = inb1 ? mm1 : 0.f;
    const float* ap1 = in + (((size_t)cz1 * G_in + cy1) * G_in + cx1) * IC;

    // Per-wave coalesced B stream: chunk j at bpt, advance 64 floats/chunk.
    const float* bpt =
        wp + (((size_t)nt * TAPS + tap) * CH) * 64 + (size_t)lane * 2;

    if (icAligned) {
      // Fast path: single aligned b64 loads for A and B fragments.
      const float* a0p = ap0 + 2 * hi;
      const float* a1p = ap1 + 2 * hi;
#pragma unroll 2
      for (int j = 0; j < CH; ++j) {
        v2f b = *(const v2f*)bpt;
        v2f a0 = *(const v2f*)a0p;
        v2f a1 = *(const v2f*)a1p;
        a0 *= am0;
        a1 *= am1;
        __builtin_prefetch(bpt + 512, 0, 1);  // global_prefetch_b8
        acc0 = wmma_f32_k4(a0, b, acc0);
        acc1 = wmma_f32_k4(a1, b, acc1);
        bpt += 64;
        a0p += 4;
        a1p += 4;
      }
    } else {
      // Channel-tail path (stem1, IC=3): mask A reads; B zeros are pre-baked.
      for (int j = 0; j < CH; ++j) {
        const int c = 4 * j + 2 * hi;
        const int c0 = c < IC ? c : IC - 1;
        const int c1 = (c + 1) < IC ? (c + 1) : IC - 1;
        const float s0 = (c < IC) ? 1.f : 0.f;
        const float s1 = ((c + 1) < IC) ? 1.f : 0.f;
        v2f b = *(const v2f*)bpt;
        v2f a0, a1;
        a0.x = ap0[c0] * (s0 * am0);
        a0.y = ap0[c1] * (s1 * am0);
        a1.x = ap1[c0] * (s0 * am1);
        a1.y = ap1[c1] * (s1 * am1);
        acc0 = wmma_f32_k4(a0, b, acc0);
        acc1 = wmma_f32_k4(a1, b, acc1);
        bpt += 64;
      }
    }
  }

  // D layout: VGPR r holds row M = r + 8*hi, column N = lane&15.
#pragma unroll
  for (int r = 0; r < 8; ++r) {
    const int va = mt * 32 + r + 8 * hi;
    if (va < Nvox) out[(size_t)va * OC + oc] = acc0[r];
    const int vb = va + 16;
    if (vb < Nvox) out[(size_t)vb * OC + oc] = acc1[r];
  }
}

// ---------------------------------------------------------------------------
// Deconv ks=2 stride=2 'VALID' (jax conv_transpose, kernel not transposed):
// out[2z+d] = sum_ic in[z,ic] * w[1-dz,1-dy,1-dx,ic,oc]
// One GEMM per output-parity tap; packed weights (TAPS=8); scatter store.
// IC is always a multiple of 4 here.
// ---------------------------------------------------------------------------
__global__ void __launch_bounds__(256) deconv2_wmma_kernel(
    const float* __restrict__ in, const float* __restrict__ wp,
    float* __restrict__ out, int G_in, int IC, int OC) {
  const int lane = threadIdx.x & 31;
  const int wid = threadIdx.x >> 5;
  const int Nvox = G_in * G_in * G_in;
  const int MT = (Nvox + 31) >> 5;
  const int NT = OC >> 4;
  const int tile = blockIdx.x * 8 + wid;
  if (tile >= MT * NT * 8) return;
  const int tap = tile & 7;
  const int t2 = tile >> 3;
  const int mt = t2 % MT;
  const int nt = t2 / MT;
  const int wtap = 7 - tap;  // kernel-tap flip

  const int mrow = lane & 15;
  const int hi = lane >> 4;
  const int oc = nt * 16 + (lane & 15);

  int v0 = mt * 32 + mrow;
  int v1 = v0 + 16;
  const float mm0 = (v0 < Nvox) ? 1.f : 0.f;
  const float mm1 = (v1 < Nvox) ? 1.f : 0.f;
  if (v0 >= Nvox) v0 = Nvox - 1;
  if (v1 >= Nvox) v1 = Nvox - 1;

  const int ICp = (IC + 3) & ~3;
  const int CH = ICp >> 2;
  v8f acc0 = {0.f, 0.f, 0.f, 0.f, 0.f, 0.f, 0.f, 0.f};
  v8f acc1 = {0.f, 0.f, 0.f, 0.f, 0.f, 0.f, 0.f, 0.f};
  const float* a0p = in + (size_t)v0 * IC + 2 * hi;
  const float* a1p = in + (size_t)v1 * IC + 2 * hi;
  const float* bpt =
      wp + (((size_t)nt * 8 + wtap) * CH) * 64 + (size_t)lane * 2;

#pragma unroll 2
  for (int j = 0; j < CH; ++j) {
    v2f b = *(const v2f*)bpt;
    v2f a0 = *(const v2f*)a0p;
    v2f a1 = *(const v2f*)a1p;
    a0 *= mm0;
    a1 *= mm1;
    acc0 = wmma_f32_k4(a0, b, acc0);
    acc1 = wmma_f32_k4(a1, b, acc1);
    bpt += 64;
    a0p += 4;
    a1p += 4;
  }

  const int G2 = G_in * 2;
  const int tz = (tap >> 2) & 1, ty = (tap >> 1) & 1, tx = tap & 1;
#pragma unroll
  for (int r = 0; r < 8; ++r) {
#pragma unroll
    for (int half = 0; half < 2; ++half) {
      const int v = mt * 32 + r + 8 * hi + 16 * half;
      if (v < Nvox) {
        const int z = v / (G_in * G_in);
        const int q = v - z * G_in * G_in;
        const int y = q / G_in;
        const int x = q - y * G_in;
        const size_t o =
            (((size_t)(2 * z + tz) * G2 + (2 * y + ty)) * G2 + (2 * x + tx));
        out[o * OC + oc] = (half == 0) ? acc0[r] : acc1[r];
      }
    }
  }
}

// ---------------------------------------------------------------------------
// BatchNorm: per-channel sum / sumsq (one block per channel, tree reduce)
// ---------------------------------------------------------------------------
__global__ void __launch_bounds__(256) bn_stats_kernel(
    const float* __restrict__ x, float* __restrict__ stats, int N, int C) {
  const int c = blockIdx.x;
  const int tid = threadIdx.x;
  float s = 0.f, q = 0.f;
  for (int v = tid; v < N; v += 256) {
    const float t = x[(size_t)v * C + c];
    s += t;
    q += t * t;
  }
  __shared__ float sh[256];
  __shared__ float sh2[256];
  sh[tid] = s;
  sh2[tid] = q;
  __syncthreads();
  for (int o = 128; o > 0; o >>= 1) {
    if (tid < o) {
      sh[tid] += sh[tid + o];
      sh2[tid] += sh2[tid + o];
    }
    __syncthreads();
  }
  if (tid == 0) {
    stats[c] = sh[0];
    stats[C + c] = sh2[0];
  }
}

// y = bn(x)*g + b  [+ add]  [relu];  in-place safe (elementwise)
__global__ void __launch_bounds__(256) bn_apply_kernel(
    const float* __restrict__ x, float* __restrict__ y,
    const float* __restrict__ stats, const float* __restrict__ g,
    const float* __restrict__ b, const float* __restrict__ add, int N, int C,
    int relu) {
  const size_t T = (size_t)N * C;
  const float invN = 1.f / (float)N;
  for (size_t i = (size_t)blockIdx.x * blockDim.x + threadIdx.x; i < T;
       i += (size_t)gridDim.x * blockDim.x) {
    const int c = (int)(i % C);
    const float mu = stats[c] * invN;
    const float var = stats[C + c] * invN - mu * mu;
    float v = (x[i] - mu) * (1.f / sqrtf(var + 1e-5f)) * g[c] + b[c];
    if (add) v += add[i];
    if (relu) v = v > 0.f ? v : 0.f;
    y[i] = v;
  }
}

// x[v,c] += cond[c]
__global__ void __launch_bounds__(256) add_vec_kernel(float* __restrict__ x,
                                                      const float* __restrict__
                                                          cond,
                                                      int N, int C) {
  const size_t T = (size_t)N * C;
  for (size_t i = (size_t)blockIdx.x * blockDim.x + threadIdx.x; i < T;
       i += (size_t)gridDim.x * blockDim.x)
    x[i] += cond[i % C];
}

// out[v, 0:C1] = a ; out[v, C1:C1+C2] = b
__global__ void __launch_bounds__(256) concat_kernel(
    const float* __restrict__ a, const float* __restrict__ b,
    float* __restrict__ o, int N, int C1, int C2) {
  const int C = C1 + C2;
  const size_t T = (size_t)N * C;
  for (size_t i = (size_t)blockIdx.x * blockDim.x + threadIdx.x; i < T;
       i += (size_t)gridDim.x * blockDim.x) {
    const size_t v = i / C;
    const int c = (int)(i - v * C);
    o[i] = (c < C1) ? a[v * C1 + c] : b[v * C2 + (c - C1)];
  }
}

// ---------------------------------------------------------------------------
// Conditioning vector: cond = MLP2(part + MLP1(temb(t)))
// MLP1: 48 ->48(leaky 0.1)-> 256 ; MLP2: 256 -> H(leaky 0.1) -> OUT
// One block, 256 threads.
// ---------------------------------------------------------------------------
__global__ void __launch_bounds__(256) cond_kernel(
    const int* __restrict__ t, const float* __restrict__ part,
    const float* __restrict__ aw1, const float* __restrict__ ab1,
    const float* __restrict__ aw2, const float* __restrict__ ab2,
    const float* __restrict__ bw1, const float* __restrict__ bb1,
    const float* __restrict__ bw2, const float* __restrict__ bb2, int H,
    int OUT, float* __restrict__ cond) {
  __shared__ float temb[48];
  __shared__ float h1[48];
  __shared__ float vv[256];
  __shared__ float h2[256];
  const int tid = threadIdx.x;
  const float tv = (float)t[0];
  if (tid < 24) {
    const float fr = expf((float)tid * (-logf(10000.f) / 23.f));
    const float a = tv * fr;
    temb[tid] = sinf(a);
    temb[tid + 24] = cosf(a);
  }
  __syncthreads();
  if (tid < 48) {
    float s = ab1[tid];
    for (int i = 0; i < 48; ++i) s += temb[i] * aw1[i * 48 + tid];
    h1[tid] = s > 0.f ? s : 0.1f * s;
  }
  __syncthreads();
  {
    float s = ab2[tid];
    for (int i = 0; i < 48; ++i) s += h1[i] * aw2[i * 256 + tid];
    vv[tid] = part[tid] + s;
  }
  __syncthreads();
  if (tid < H) {
    float s = bb1[tid];
    for (int i = 0; i < 256; ++i) s += vv[i] * bw1[i * H + tid];
    h2[tid] = s > 0.f ? s : 0.1f * s;
  }
  __syncthreads();
  if (tid < OUT) {
    float s = bb2[tid];
    for (int i = 0; i < H; ++i) s += h2[i] * bw2[i * OUT + tid];
    cond[tid] = s;
  }
}

// ---------------------------------------------------------------------------
// Per-voxel head: 48 -> 40 (leaky 0.1) -> 3
// ---------------------------------------------------------------------------
__global__ void __launch_bounds__(128) head_kernel(
    const float* __restrict__ x, const float* __restrict__ w1,
    const float* __restrict__ b1, const float* __restrict__ w2,
    const float* __restrict__ b2, float* __restrict__ out, int N) {
  __shared__ float sw1[48 * 40];
  __shared__ float sb1[40];
  __shared__ float sw2[40 * 3];
  __shared__ float sb2[3];
  for (int i = threadIdx.x; i < 48 * 40; i += 128) sw1[i] = w1[i];
  for (int i = threadIdx.x; i < 40; i += 128) sb1[i] = b1[i];
  for (int i = threadIdx.x; i < 120; i += 128) sw2[i] = w2[i];
  if (threadIdx.x < 3) sb2[threadIdx.x] = b2[threadIdx.x];
  __syncthreads();
  const int v = blockIdx.x * 128 + threadIdx.x;
  if (v >= N) return;
  float xi[48];
  for (int i = 0; i < 48; ++i) xi[i] = x[(size_t)v * 48 + i];
  float o0 = sb2[0], o1 = sb2[1], o2 = sb2[2];
  for (int j = 0; j < 40; ++j) {
    float h = sb1[j];
    for (int i = 0; i < 48; ++i) h += xi[i] * sw1[i * 40 + j];
    h = h > 0.f ? h : 0.1f * h;
    o0 += h * sw2[j * 3 + 0];
    o1 += h * sw2[j * 3 + 1];
    o2 += h * sw2[j * 3 + 2];
  }
  out[(size_t)v * 3 + 0] = o0;
  out[(size_t)v * 3 + 1] = o1;
  out[(size_t)v * 3 + 2] = o2;
}

// ---------------------------------------------------------------------------
// Host-side orchestration
// ---------------------------------------------------------------------------
struct ConvP {
  const float *b, *g, *w;
};
struct ResP {
  const float *b1, *b2, *bd, *g1, *g2, *gd, *w1, *w2, *wd;
  int has_wd;
};
struct MlpP {
  const float *b1, *b2, *w1, *w2;
};

// Packed-weight scratch (set once in kernel_launch before any conv launch).
static float* g_wpack = nullptr;

static void launch_conv(hipStream_t s, const float* in, const float* w,
                        float* out, int gin, int gout, int ic, int oc, int ks,
                        int stride, int pad) {
  const int taps = ks * ks * ks;
  const int icp = (ic + 3) & ~3;
  const size_t wt = (size_t)oc * taps * icp;
  int pblk = (int)((wt + 255) / 256);
  if (pblk > 2048) pblk = 2048;
  pack_w_kernel<<<pblk, 256, 0, s>>>(w, g_wpack, taps, ic, oc, icp);
  const int nvox = gout * gout * gout;
  const int tiles = ((nvox + 31) / 32) * (oc / 16);
  const int blocks = (tiles + 7) / 8;
  conv3d_wmma_kernel<<<blocks, 256, 0, s>>>(in, g_wpack, out, gin, gout, ic,
                                            oc, ks, stride, pad);
}

static void launch_deconv(hipStream_t s, const float* in, const float* w,
                          float* out, int gin, int ic, int oc) {
  const int icp = (ic + 3) & ~3;
  const size_t wt = (size_t)oc * 8 * icp;
  int pblk = (int)((wt + 255) / 256);
  if (pblk > 2048) pblk = 2048;
  pack_w_kernel<<<pblk, 256, 0, s>>>(w, g_wpack, 8, ic, oc, icp);
  const int nvox = gin * gin * gin;
  const int tiles = ((nvox + 31) / 32) * (oc / 16) * 8;
  const int blocks = (tiles + 7) / 8;
  deconv2_wmma_kernel<<<blocks, 256, 0, s>>>(in, g_wpack, out, gin, ic, oc);
}

static void launch_bn(hipStream_t s, float* stats, const float* x, float* y,
                      const float* g, const float* b, int N, int C, int relu,
                      const float* add) {
  bn_stats_kernel<<<C, 256, 0, s>>>(x, stats, N, C);
  size_t T = (size_t)N * C;
  int blk = (int)((T + 255) / 256);
  if (blk > 4096) blk = 4096;
  bn_apply_kernel<<<blk, 256, 0, s>>>(x, y, stats, g, b, add, N, C, relu);
}

// Residual block. x consumed; fb1/fb2 are free scratch buffers (distinct from
// x). Returns buffer holding result (x's buffer if has_wd, else fb2).
static float* run_res(hipStream_t s, const ResP& p, const float* x, float* fb1,
                      float* fb2, int g, int ic, int oc, float* stats) {
  const int N = g * g * g;
  if (p.has_wd) {
    launch_conv(s, x, p.wd, fb2, g, g, ic, oc, 1, 1, 0);
    launch_bn(s, stats, fb2, fb2, p.gd, p.bd, N, oc, 0, nullptr);
    launch_conv(s, x, p.w1, fb1, g, g, ic, oc, 3, 1, 1);
    launch_bn(s, stats, fb1, fb1, p.g1, p.b1, N, oc, 1, nullptr);
    float* out = (float*)x;  // x dead after both reads above
    launch_conv(s, fb1, p.w2, out, g, g, oc, oc, 3, 1, 1);
    launch_bn(s, stats, out, out, p.g2, p.b2, N, oc, 1, fb2);
    return out;
  } else {
    launch_conv(s, x, p.w1, fb1, g, g, ic, oc, 3, 1, 1);
    launch_bn(s, stats, fb1, fb1, p.g1, p.b1, N, oc, 1, nullptr);
    launch_conv(s, fb1, p.w2, fb2, g, g, oc, oc, 3, 1, 1);
    launch_bn(s, stats, fb2, fb2, p.g2, p.b2, N, oc, 1, x);  // + identity
    return fb2;
  }
}

extern "C" void kernel_launch(void* const* d_in, const int* in_sizes, int n_in,
                              void* d_out, int out_size, void* d_ws,
                              size_t ws_size, hipStream_t stream) {
  (void)in_sizes;
  (void)out_size;
  (void)ws_size;
  static const int CSa[13] = {16, 16, 32, 64, 128, 256, 256,
                              256, 256, 128, 64, 48, 48};

  // ---- param un-flattening (jax pytree: nested dicts in sorted-key order,
  //      after the top-level x, part, t) ----
  int ix = 3;
  auto F = [&]() -> const float* {
    const int k = (ix < n_in) ? ix : (n_in - 1);
    ++ix;
    return (const float*)d_in[k];
  };
  auto RES = [&](ResP& r, bool wd) {
    r.has_wd = wd ? 1 : 0;
    r.b1 = F();
    r.b2 = F();
    r.bd = wd ? F() : nullptr;
    r.g1 = F();
    r.g2 = F();
    r.gd = wd ? F() : nullptr;
    r.w1 = F();
    r.w2 = F();
    r.wd = wd ? F() : nullptr;
  };
  auto MLP = [&](MlpP& m) {
    m.b1 = F();
    m.b2 = F();
    m.w1 = F();
    m.w2 = F();
  };

  MlpP last;
  MLP(last);
  MlpP lat_st[7], lat_up[7];
  for (int i = 1; i <= 6; ++i) MLP(lat_st[i]);
  for (int j = 1; j <= 6; ++j) MLP(lat_up[j]);
  ConvP down[7];
  ResP st_r1[7], st_r2[7];
  MlpP st_temp[7];
  for (int i = 1; i <= 6; ++i) {
    down[i].b = F();
    down[i].g = F();
    down[i].w = F();
    RES(st_r1[i], CSa[i - 1] != CSa[i]);
    RES(st_r2[i], false);
    MLP(st_temp[i]);
  }
  ConvP stem1, stem2;
  stem1.b = F();
  stem1.g = F();
  stem1.w = F();
  stem2.b = F();
  stem2.g = F();
  stem2.w = F();
  ConvP de[7];
  ResP up_r1[7], up_r2[7];
  MlpP up_temp[7];
  for (int j = 1; j <= 6; ++j) {
    de[j].b = F();
    de[j].g = F();
    de[j].w = F();
    RES(up_r1[j], true);
    RES(up_r2[j], false);
    MLP(up_temp[j]);
  }

  const float* x_in = (const float*)d_in[0];
  const float* part = (const float*)d_in[1];
  const int* tptr = (const int*)d_in[2];

  // ---- workspace arena (floats) ----
  const size_t VOX64 = 262144;     // 64^3
  const size_t BIGF = VOX64 * 64;  // 16.7M floats per arena
  float* ws = (float*)d_ws;
  float* bufs[3] = {ws, ws + BIGF, ws + 2 * BIGF};
  size_t off = 3 * BIGF;
  static const int skN[7] = {262144, 32768, 4096, 512, 64, 8, 1};
  static const int skC[7] = {16, 16, 32, 64, 128, 256, 256};
  float* SK[7];
  for (int k = 0; k < 7; ++k) {
    SK[k] = ws + off;
    off += (size_t)skN[k] * skC[k];
  }
  float* COND = ws + off;
  off += 256;
  float* STATS = ws + off;
  off += 1024;
  g_wpack = ws + off;  // max packed weight: 256*27*512 = 3,538,944 floats
  off += 3538944;

  auto bidx = [&](float* p) -> int {
    return p == bufs[0] ? 0 : (p == bufs[1] ? 1 : 2);
  };
  auto grid_for = [](size_t T) {
    int blk = (int)((T + 255) / 256);
    return blk > 4096 ? 4096 : blk;
  };

  // ---- stems (64^3) ----
  launch_conv(stream, x_in, stem1.w, bufs[0], 64, 64, 3, 16, 3, 1, 1);
  launch_bn(stream, STATS, bufs[0], bufs[0], stem1.g, stem1.b, (int)VOX64, 16,
            1, nullptr);
  launch_conv(stream, bufs[0], stem2.w, bufs[1], 64, 64, 16, 16, 3, 1, 1);
  launch_bn(stream, STATS, bufs[1], bufs[1], stem2.g, stem2.b, (int)VOX64, 16,
            1, nullptr);
  (void)hipMemcpyAsync(SK[0], bufs[1], VOX64 * 16 * sizeof(float),
                       hipMemcpyDeviceToDevice, stream);

  int hi = 1;
  int g = 64;

  // ---- encoder ----
  for (int i = 1; i <= 6; ++i) {
    const int ic = CSa[i - 1], oc = CSa[i];
    cond_kernel<<<1, 256, 0, stream>>>(
        tptr, part, st_temp[i].w1, st_temp[i].b1, st_temp[i].w2, st_temp[i].b2,
        lat_st[i].w1, lat_st[i].b1, lat_st[i].w2, lat_st[i].b2, 256, ic, COND);
    int N = g * g * g;
    add_vec_kernel<<<grid_for((size_t)N * ic), 256, 0, stream>>>(bufs[hi],
                                                                 COND, N, ic);
    const int f1 = (hi + 1) % 3, f2 = (hi + 2) % 3;
    launch_conv(stream, bufs[hi], down[i].w, bufs[f1], g, g / 2, ic, ic, 2, 2,
                0);
    g /= 2;
    N = g * g * g;
    launch_bn(stream, STATS, bufs[f1], bufs[f1], down[i].g, down[i].b, N, ic,
              1, nullptr);
    float* r1o = run_res(stream, st_r1[i], bufs[f1], bufs[hi], bufs[f2], g, ic,
                         oc, STATS);
    const int r1i = bidx(r1o);
    const int a = (r1i + 1) % 3, b = (r1i + 2) % 3;
    float* r2o = run_res(stream, st_r2[i], bufs[r1i], bufs[a], bufs[b], g, oc,
                         oc, STATS);
    hi = bidx(r2o);
    (void)hipMemcpyAsync(SK[i], bufs[hi], (size_t)N * oc * sizeof(float),
                         hipMemcpyDeviceToDevice, stream);
  }

  // ---- decoder ----
  static const int icup[7] = {0, 256, 256, 256, 128, 64, 48};
  static const int ocup[7] = {0, 256, 256, 128, 64, 48, 48};
  static const int skc[7] = {0, 256, 128, 64, 32, 16, 16};
  for (int j = 1; j <= 6; ++j) {
    const int ic = icup[j], oc = ocup[j], sk = skc[j];
    cond_kernel<<<1, 256, 0, stream>>>(
        tptr, part, up_temp[j].w1, up_temp[j].b1, up_temp[j].w2, up_temp[j].b2,
        lat_up[j].w1, lat_up[j].b1, lat_up[j].w2, lat_up[j].b2, ic, ic, COND);
    int N = g * g * g;
    add_vec_kernel<<<grid_for((size_t)N * ic), 256, 0, stream>>>(bufs[hi],
                                                                 COND, N, ic);
    const int f1 = (hi + 1) % 3, f2 = (hi + 2) % 3;
    launch_deconv(stream, bufs[hi], de[j].w, bufs[f1], g, ic, oc);
    g *= 2;
    N = g * g * g;
    launch_bn(stream, STATS, bufs[f1], bufs[f1], de[j].g, de[j].b, N, oc, 1,
              nullptr);
    concat_kernel<<<grid_for((size_t)N * (oc + sk)), 256, 0, stream>>>(
        bufs[f1], SK[6 - j], bufs[f2], N, oc, sk);
    float* r1o = run_res(stream, up_r1[j], bufs[f2], bufs[f1], bufs[hi], g,
                         oc + sk, oc, STATS);
    const int r1i = bidx(r1o);
    const int a = (r1i + 1) % 3, b = (r1i + 2) % 3;
    float* r2o = run_res(stream, up_r2[j], bufs[r1i], bufs[a], bufs[b], g, oc,
                         oc, STATS);
    hi = bidx(r2o);
  }

  // ---- per-voxel head ----
  head_kernel<<<(int)((VOX64 + 127) / 128), 128, 0, stream>>>(
      bufs[hi], last.w1, last.b1, last.w2, last.b2, (float*)d_out,
      (int)VOX64);
}